// PointTransformer_cond_62878321214064
// MI455X (gfx1250) — compile-verified
//
#include <hip/hip_runtime.h>
#include <math.h>

// Problem constants (match reference)
constexpr int B_ = 2, N_ = 4096, C_ = 128, L_ = 4, K_ = 16;
constexpr int BN_ = B_ * N_;        // 8192
constexpr int BNK_ = BN_ * K_;      // 131072
constexpr float EPS_ = 1e-5f;

typedef __attribute__((ext_vector_type(2))) float v2f;
typedef __attribute__((ext_vector_type(8))) float v8f;

// ---------------------------------------------------------------------------
// WMMA fp32 GEMM:  D[M,Nout] = act(A[M,Kd] @ W[Kd,Nout] + bias) (+ res)
// Register-blocked: block = 64-row M super-tile x 128-col N super-tile.
// 8 waves, wave w owns n-tile w (16 cols). K swept in 128-chunks:
//   - A chunk (64x128) staged in LDS (stride 130 -> conflict-free ds_load_b64)
//   - B frags for the chunk held in 64 VGPRs (32 x v2f), loaded ONCE
//   - wave sweeps 4 m-tiles => 128 v_wmma_f32_16x16x4_f32 per chunk,
//     4 independent accumulator chains.
// act: 0=none, 1=relu, 2=gelu(exact erf).  All dims divide exactly -> EXEC
// all-ones for every WMMA (ISA requirement).
// ---------------------------------------------------------------------------
__global__ __launch_bounds__(256) void gemm_wmma(
    const float* __restrict__ A, const float* __restrict__ W,
    const float* __restrict__ bias, const float* __restrict__ res,
    float* __restrict__ Dst, int M, int Kd, int Nout, int act)
{
  __shared__ float As[64 * 130];          // 33 KB, +2 float pad per row
  const int tid  = threadIdx.x;
  const int m0   = blockIdx.x * 64;
  const int wave = tid >> 5;
  const int lane = tid & 31;              // wave32
  const int half = lane >> 4;             // selects K pair {0,1} vs {2,3}
  const int lm   = lane & 15;
  const int n0   = blockIdx.y * 128 + wave * 16;

  v8f acc[4];
#pragma unroll
  for (int mt = 0; mt < 4; ++mt) acc[mt] = {};

  for (int kc = 0; kc < Kd; kc += 128) {
    // ---- stage A[64][kc:kc+128] into LDS (coalesced) ----
    for (int i = tid; i < 64 * 128; i += 256) {
      int r = i >> 7, k = i & 127;
      As[r * 130 + k] = A[(size_t)(m0 + r) * Kd + kc + k];
    }
    // speculative prefetch of the next K chunk (global_prefetch_b8)
    if (kc + 128 < Kd)
      __builtin_prefetch(&A[(size_t)(m0 + (tid >> 2)) * Kd + kc + 128 + (tid & 3) * 32], 0, 1);
    __syncthreads();

    // ---- B fragments for this wave's 16-col panel, whole chunk ----
    v2f bf[32];
    const float* wp = W + ((size_t)kc + 2 * half) * Nout + n0 + lm;
#pragma unroll
    for (int kk = 0; kk < 32; ++kk) {
      bf[kk].x = wp[(size_t)(4 * kk) * Nout];
      bf[kk].y = wp[(size_t)(4 * kk + 1) * Nout];
    }

    // ---- 4 m-tiles x 32 k-steps of WMMA ----
#pragma unroll
    for (int mt = 0; mt < 4; ++mt) {
      const float* ap = As + (mt * 16 + lm) * 130 + 2 * half;
#pragma unroll
      for (int kk = 0; kk < 32; ++kk) {
        v2f a = *(const v2f*)(ap + 4 * kk);
        acc[mt] = __builtin_amdgcn_wmma_f32_16x16x4_f32(false, a, false, bf[kk],
                                                        (short)0, acc[mt], false, false);
      }
    }
    __syncthreads();
  }

  // ---- epilogue: bias, activation, residual ----
  const int col = n0 + lm;
  const float bv = bias ? bias[col] : 0.f;
#pragma unroll
  for (int mt = 0; mt < 4; ++mt) {
#pragma unroll
    for (int g = 0; g < 8; ++g) {
      int row = m0 + mt * 16 + g + 8 * half;
      float v = acc[mt][g] + bv;
      if (act == 1)      v = fmaxf(v, 0.f);
      else if (act == 2) v = 0.5f * v * (1.f + erff(v * 0.70710678118654752f));
      size_t o = (size_t)row * Nout + col;
      if (res) v += res[o];
      Dst[o] = v;
    }
  }
}

// ---------------------------------------------------------------------------
// kNN: one thread per query point, register-resident sorted top-16
// (unrolled compare-swap ladder => stays in VGPRs), LDS-tiled candidates.
// ---------------------------------------------------------------------------
__global__ __launch_bounds__(64) void knn_kernel(const float* __restrict__ pos,
                                                 int* __restrict__ idx)
{
  __shared__ float sp[128 * 3];
  const int q = blockIdx.x * 64 + threadIdx.x;   // all threads in block share b
  const int b = q / N_;
  const int n = q - b * N_;
  const float px = pos[(size_t)(b * N_ + n) * 3 + 0];
  const float py = pos[(size_t)(b * N_ + n) * 3 + 1];
  const float pz = pos[(size_t)(b * N_ + n) * 3 + 2];

  float bd[K_]; int bi[K_];
#pragma unroll
  for (int i = 0; i < K_; ++i) { bd[i] = 3.4e38f; bi[i] = 0; }

  for (int m0 = 0; m0 < N_; m0 += 128) {
    __syncthreads();
    for (int i = threadIdx.x; i < 128 * 3; i += 64)
      sp[i] = pos[(size_t)(b * N_ + m0) * 3 + i];
    __syncthreads();
    for (int mm = 0; mm < 128; ++mm) {
      float dx = px - sp[mm * 3 + 0];
      float dy = py - sp[mm * 3 + 1];
      float dz = pz - sp[mm * 3 + 2];
      float d2 = dx * dx + dy * dy + dz * dz;
      if (d2 < bd[K_ - 1]) {
        float dv = d2; int di = m0 + mm;
#pragma unroll
        for (int j = 0; j < K_; ++j) {
          if (dv < bd[j]) {
            float tf = bd[j]; int ti = bi[j];
            bd[j] = dv; bi[j] = di; dv = tf; di = ti;
          }
        }
      }
    }
  }
#pragma unroll
  for (int k = 0; k < K_; ++k) idx[(size_t)(b * N_ + n) * K_ + k] = bi[k];
}

// ---------------------------------------------------------------------------
// AdaLN conditioning: h = LN(cond@w1+b1)*g+b ; out = h@w2+b2   -> ss[B, 2C]
// ---------------------------------------------------------------------------
__global__ __launch_bounds__(256) void adapt_kernel(
    const float* __restrict__ cond,
    const float* __restrict__ w1, const float* __restrict__ b1,
    const float* __restrict__ g,  const float* __restrict__ bb,
    const float* __restrict__ w2, const float* __restrict__ b2,
    float* __restrict__ ss)
{
  __shared__ float sh[256];
  __shared__ float red[256];
  const int b = blockIdx.x, c = threadIdx.x;
  float h = b1[c];
#pragma unroll
  for (int d = 0; d < 3; ++d) h += cond[b * 3 + d] * w1[d * 256 + c];
  red[c] = h; __syncthreads();
  for (int s = 128; s > 0; s >>= 1) { if (c < s) red[c] += red[c + s]; __syncthreads(); }
  const float mu = red[0] * (1.f / 256.f); __syncthreads();
  const float dm = h - mu;
  red[c] = dm * dm; __syncthreads();
  for (int s = 128; s > 0; s >>= 1) { if (c < s) red[c] += red[c + s]; __syncthreads(); }
  const float var = red[0] * (1.f / 256.f); __syncthreads();
  sh[c] = dm * rsqrtf(var + EPS_) * g[c] + bb[c];
  __syncthreads();
  float o = b2[c];
  for (int j = 0; j < 256; ++j) o += sh[j] * w2[j * 256 + c];
  ss[b * 256 + c] = o;
}

// xm = x * (1 + scale) + shift
__global__ void modulate_kernel(const float* __restrict__ x,
                                const float* __restrict__ ss,
                                float* __restrict__ xm)
{
  const int e = blockIdx.x * 256 + threadIdx.x;     // < BN*C
  const int c = e & (C_ - 1);
  const int b = e / (N_ * C_);
  xm[e] = x[e] * (1.f + ss[b * 2 * C_ + C_ + c]) + ss[b * 2 * C_ + c];
}

// pe1 = relu(pos_diff @ pos_w1 + pos_b1), pos_diff gathered on the fly (Kd=3)
__global__ void pe1_kernel(const float* __restrict__ pos, const int* __restrict__ idx,
                           const float* __restrict__ w1, const float* __restrict__ b1,
                           float* __restrict__ pe1)
{
  const int e = blockIdx.x * 256 + threadIdx.x;     // < BNK*C
  const int c = e & (C_ - 1);
  const int bnk = e >> 7;                           // C_=128
  const int bn  = bnk >> 4;                         // K_=16
  const int b   = bn >> 12;                         // N_=4096
  const int j   = idx[bnk];
  float acc = b1[c];
#pragma unroll
  for (int d = 0; d < 3; ++d) {
    float pd = pos[(size_t)bn * 3 + d] - pos[(size_t)(b * N_ + j) * 3 + d];
    acc += pd * w1[d * C_ + c];
  }
  pe1[e] = fmaxf(acc, 0.f);
}

// rel = gather(k) - q + pe
__global__ void rel_kernel(const float* __restrict__ kv, const float* __restrict__ q,
                           const float* __restrict__ pe, const int* __restrict__ idx,
                           float* __restrict__ rel)
{
  const int e = blockIdx.x * 256 + threadIdx.x;     // < BNK*C
  const int c = e & (C_ - 1);
  const int bnk = e >> 7;
  const int bn  = bnk >> 4;
  const int b   = bn >> 12;
  const int j   = idx[bnk];
  rel[e] = kv[(size_t)(b * N_ + j) * (2 * C_) + c] - q[(size_t)bn * C_ + c] + pe[e];
}

// softmax over K (per channel) fused with V-gather + weighted aggregation
__global__ __launch_bounds__(128) void smax_agg_kernel(
    const float* __restrict__ a, const float* __restrict__ kv,
    const float* __restrict__ pe, const int* __restrict__ idx,
    float* __restrict__ agg)
{
  __shared__ int sj[K_];
  const int bn = blockIdx.x;
  const int c  = threadIdx.x;
  if (c < K_) sj[c] = idx[bn * K_ + c];
  __syncthreads();
  const int b = bn >> 12;
  float av[K_];
  float mx = -3.4e38f;
#pragma unroll
  for (int k = 0; k < K_; ++k) {
    av[k] = a[((size_t)bn * K_ + k) * C_ + c];
    mx = fmaxf(mx, av[k]);
  }
  float s = 0.f;
#pragma unroll
  for (int k = 0; k < K_; ++k) { av[k] = expf(av[k] - mx); s += av[k]; }
  const float inv = 1.f / s;
  float o = 0.f;
#pragma unroll
  for (int k = 0; k < K_; ++k) {
    float val = kv[(size_t)(b * N_ + sj[k]) * (2 * C_) + C_ + c]
              + pe[((size_t)bn * K_ + k) * C_ + c];
    o += av[k] * inv * val;
  }
  agg[(size_t)bn * C_ + c] = o;
}

// LayerNorm over F, optional exact-GELU after, optional residual add
template <int F>
__global__ void ln_kernel(const float* __restrict__ xin,
                          const float* __restrict__ g, const float* __restrict__ bb,
                          const float* __restrict__ res, float* __restrict__ out,
                          int dogelu)
{
  __shared__ float red[F];
  const int r = blockIdx.x, c = threadIdx.x;
  const float v = xin[(size_t)r * F + c];
  red[c] = v; __syncthreads();
  for (int s = F / 2; s > 0; s >>= 1) { if (c < s) red[c] += red[c + s]; __syncthreads(); }
  const float mu = red[0] * (1.f / F); __syncthreads();
  const float dm = v - mu;
  red[c] = dm * dm; __syncthreads();
  for (int s = F / 2; s > 0; s >>= 1) { if (c < s) red[c] += red[c + s]; __syncthreads(); }
  const float var = red[0] * (1.f / F);
  float y = dm * rsqrtf(var + EPS_) * g[c] + bb[c];
  if (dogelu) y = 0.5f * y * (1.f + erff(y * 0.70710678118654752f));
  if (res) y += res[(size_t)r * F + c];
  out[(size_t)r * F + c] = y;
}

// x = pos @ proj_in_w + proj_in_b
__global__ void proj_in_kernel(const float* __restrict__ pos, const float* __restrict__ w,
                               const float* __restrict__ b, float* __restrict__ x)
{
  const int e = blockIdx.x * 256 + threadIdx.x;     // < BN*C
  const int c = e & (C_ - 1);
  const int bn = e >> 7;
  float acc = b[c];
#pragma unroll
  for (int d = 0; d < 3; ++d) acc += pos[(size_t)bn * 3 + d] * w[d * C_ + c];
  x[e] = acc;
}

// out = x @ proj_out_w + proj_out_b   (Nout = 1)
__global__ __launch_bounds__(128) void proj_out_kernel(const float* __restrict__ x,
                                                       const float* __restrict__ w,
                                                       const float* __restrict__ b,
                                                       float* __restrict__ out)
{
  __shared__ float red[128];
  const int bn = blockIdx.x, c = threadIdx.x;
  red[c] = x[(size_t)bn * C_ + c] * w[c];
  __syncthreads();
  for (int s = 64; s > 0; s >>= 1) { if (c < s) red[c] += red[c + s]; __syncthreads(); }
  if (c == 0) out[bn] = red[0] + b[0];
}

// ---------------------------------------------------------------------------
extern "C" void kernel_launch(void* const* d_in, const int* in_sizes, int n_in,
                              void* d_out, int out_size, void* d_ws, size_t ws_size,
                              hipStream_t stream)
{
  (void)in_sizes; (void)n_in; (void)out_size; (void)ws_size;

  const float* pos        = (const float*)d_in[0];
  const float* cond       = (const float*)d_in[1];
  const float* proj_in_w  = (const float*)d_in[2];
  const float* proj_in_b  = (const float*)d_in[3];
  const float* proj_out_w = (const float*)d_in[4];
  const float* proj_out_b = (const float*)d_in[5];
  // layers dict, insertion order:
  const float* adapt_w1  = (const float*)d_in[6];
  const float* adapt_b1  = (const float*)d_in[7];
  const float* adapt_ln_g= (const float*)d_in[8];
  const float* adapt_ln_b= (const float*)d_in[9];
  const float* adapt_w2  = (const float*)d_in[10];
  const float* adapt_b2  = (const float*)d_in[11];
  const float* q_w   = (const float*)d_in[12];
  const float* q_b   = (const float*)d_in[13];
  const float* kv_w  = (const float*)d_in[14];
  const float* kv_b  = (const float*)d_in[15];
  const float* pos_w1= (const float*)d_in[16];
  const float* pos_b1= (const float*)d_in[17];
  const float* pos_w2= (const float*)d_in[18];
  const float* pos_b2= (const float*)d_in[19];
  const float* attn_w1=(const float*)d_in[20];
  const float* attn_b1=(const float*)d_in[21];
  const float* attn_w2=(const float*)d_in[22];
  const float* attn_b2=(const float*)d_in[23];
  const float* out_w = (const float*)d_in[24];
  const float* out_b = (const float*)d_in[25];
  const float* ffn_w1= (const float*)d_in[26];
  const float* ffn_b1= (const float*)d_in[27];
  const float* ln1_g = (const float*)d_in[28];
  const float* ln1_b = (const float*)d_in[29];
  const float* ffn_w2= (const float*)d_in[30];
  const float* ffn_b2= (const float*)d_in[31];
  const float* ln2_g = (const float*)d_in[32];
  const float* ln2_b = (const float*)d_in[33];

  // ---- workspace partition (bytes, 256B aligned) ----
  char* p = (char*)d_ws;
  auto alloc = [&](size_t bytes) -> void* {
    void* r = (void*)p;
    p += (bytes + 255) & ~(size_t)255;
    return r;
  };
  int*   idx = (int*)  alloc((size_t)BNK_ * 4);
  float* x   = (float*)alloc((size_t)BN_ * C_ * 4);
  float* xm  = (float*)alloc((size_t)BN_ * C_ * 4);
  float* x1  = (float*)alloc((size_t)BN_ * C_ * 4);
  float* qb  = (float*)alloc((size_t)BN_ * C_ * 4);
  float* kvb = (float*)alloc((size_t)BN_ * 2 * C_ * 4);
  float* agg = (float*)alloc((size_t)BN_ * C_ * 4);
  float* f1  = (float*)alloc((size_t)BN_ * 4 * C_ * 4);
  float* f2  = (float*)alloc((size_t)BN_ * C_ * 4);
  float* ss  = (float*)alloc((size_t)B_ * 2 * C_ * 4);
  float* pe1 = (float*)alloc((size_t)BNK_ * C_ * 4);   // reused as t1
  float* pe  = (float*)alloc((size_t)BNK_ * C_ * 4);
  float* rel = (float*)alloc((size_t)BNK_ * C_ * 4);   // reused as a

  auto gemm = [&](const float* A, const float* W, const float* bias, const float* res,
                  float* Dst, int M, int Kd, int Nout, int act) {
    dim3 grid(M / 64, Nout / 128);
    gemm_wmma<<<grid, 256, 0, stream>>>(A, W, bias, res, Dst, M, Kd, Nout, act);
  };

  knn_kernel<<<BN_ / 64, 64, 0, stream>>>(pos, idx);
  proj_in_kernel<<<BN_ * C_ / 256, 256, 0, stream>>>(pos, proj_in_w, proj_in_b, x);

  for (int l = 0; l < L_; ++l) {
    adapt_kernel<<<B_, 256, 0, stream>>>(cond,
        adapt_w1 + (size_t)l * 3 * 256, adapt_b1 + (size_t)l * 256,
        adapt_ln_g + (size_t)l * 256,   adapt_ln_b + (size_t)l * 256,
        adapt_w2 + (size_t)l * 256 * 256, adapt_b2 + (size_t)l * 256, ss);
    modulate_kernel<<<BN_ * C_ / 256, 256, 0, stream>>>(x, ss, xm);

    gemm(xm, q_w  + (size_t)l * C_ * C_,     q_b  + (size_t)l * C_,     nullptr, qb,  BN_, C_, C_, 0);
    gemm(xm, kv_w + (size_t)l * C_ * 2 * C_, kv_b + (size_t)l * 2 * C_, nullptr, kvb, BN_, C_, 2 * C_, 0);

    pe1_kernel<<<BNK_ * C_ / 256, 256, 0, stream>>>(pos, idx,
        pos_w1 + (size_t)l * 3 * C_, pos_b1 + (size_t)l * C_, pe1);
    gemm(pe1, pos_w2 + (size_t)l * C_ * C_, pos_b2 + (size_t)l * C_, nullptr, pe, BNK_, C_, C_, 0);

    rel_kernel<<<BNK_ * C_ / 256, 256, 0, stream>>>(kvb, qb, pe, idx, rel);

    gemm(rel, attn_w1 + (size_t)l * C_ * C_, attn_b1 + (size_t)l * C_, nullptr, pe1, BNK_, C_, C_, 1); // t1 = relu(...)
    gemm(pe1, attn_w2 + (size_t)l * C_ * C_, attn_b2 + (size_t)l * C_, nullptr, rel, BNK_, C_, C_, 0); // a

    smax_agg_kernel<<<BN_, 128, 0, stream>>>(rel, kvb, pe, idx, agg);

    gemm(agg, out_w + (size_t)l * C_ * C_, out_b + (size_t)l * C_, xm, x1, BN_, C_, C_, 0); // x1 = xm + attn

    gemm(x1, ffn_w1 + (size_t)l * C_ * 4 * C_, ffn_b1 + (size_t)l * 4 * C_, nullptr, f1, BN_, C_, 4 * C_, 0);
    ln_kernel<512><<<BN_, 512, 0, stream>>>(f1, ln1_g + (size_t)l * 4 * C_,
                                            ln1_b + (size_t)l * 4 * C_, nullptr, f1, 1); // LN + GELU in place
    gemm(f1, ffn_w2 + (size_t)l * 4 * C_ * C_, ffn_b2 + (size_t)l * C_, nullptr, f2, BN_, 4 * C_, C_, 0);
    ln_kernel<128><<<BN_, 128, 0, stream>>>(f2, ln2_g + (size_t)l * C_,
                                            ln2_b + (size_t)l * C_, x1, x, 0);           // x = x1 + LN(f2)
  }

  proj_out_kernel<<<BN_, 128, 0, stream>>>(x, proj_out_w, proj_out_b, (float*)d_out);
}